// HolographicCompressor_60309930771063
// MI455X (gfx1250) — compile-verified
//
#include <hip/hip_runtime.h>
#include <hip/hip_bf16.h>

// Shapes from the reference
#define BATCH   4
#define SEQ     4096
#define DM      1024      // D_MODEL
#define NS      128       // STATE
#define CHUNK   256       // scan chunk length
#define NCHUNK  (SEQ / CHUNK)   // 16

typedef __attribute__((ext_vector_type(2))) float v2f;
typedef __attribute__((ext_vector_type(8))) float v8f;

// ---------------------------------------------------------------------------
// K1: embedding gather  x[b,s,:] = embedding[tokens[b,s], :]
// grid = B*S rows, 256 threads, float4 per thread (256*4 = 1024 = DM)
// ---------------------------------------------------------------------------
__global__ void gather_embed_kernel(const int* __restrict__ tokens,
                                    const float* __restrict__ emb,
                                    float* __restrict__ X) {
    const long row = blockIdx.x;                 // b*SEQ + s
    const int  tok = tokens[row];
    const float4* __restrict__ src = (const float4*)(emb + (long)tok * DM);
    float4* __restrict__ dst = (float4*)(X + row * DM);
    dst[threadIdx.x] = src[threadIdx.x];
}

// ---------------------------------------------------------------------------
// K2: Bt[m, n] = sum_k X[m, k] * B_w[n, k]      (m = b*SEQ+s)
// f32 WMMA 16x16x4: one wave per 16x16 output tile.
// Block = 256 threads = 8 waves; wave w covers n-tile w (N=128 -> 8 tiles),
// blockIdx.x covers the M tile (16384/16 = 1024 blocks).
//
// Fragment layout per CDNA5 ISA 7.12.2 (32-bit A, 16x4):
//   v0/v1 hold K = {0,1} for lanes 0-15 and K = {2,3} for lanes 16-31,
//   M = lane % 16.  B (4x16) mirrored: N = lane%16, K pair by lane half.
// C/D: VGPR r -> M = r + 8*(lane>=16), N = lane%16.
// ---------------------------------------------------------------------------
__global__ void bproj_wmma_kernel(const float* __restrict__ X,
                                  const float* __restrict__ Bw,
                                  float* __restrict__ Bt) {
    const int wave = threadIdx.x >> 5;      // 0..7 -> n tile
    const int lane = threadIdx.x & 31;
    const int l15  = lane & 15;
    const int hi   = lane >> 4;             // 0 or 1
    const int m0   = blockIdx.x * 16;
    const int n0   = wave * 16;

    const float* __restrict__ arow = X  + (long)(m0 + l15) * DM + hi * 2;
    const float* __restrict__ brow = Bw + (long)(n0 + l15) * DM + hi * 2;

    v8f acc = {};
    for (int k = 0; k < DM; k += 4) {
        v2f a; a.x = arow[k]; a.y = arow[k + 1];
        v2f b; b.x = brow[k]; b.y = brow[k + 1];
        acc = __builtin_amdgcn_wmma_f32_16x16x4_f32(
            /*neg_a=*/false, a, /*neg_b=*/false, b,
            /*c_mod=*/(short)0, acc, /*reuse_a=*/false, /*reuse_b=*/false);
    }

    float* __restrict__ out = Bt + (long)(m0 + hi * 8) * NS + n0 + l15;
#pragma unroll
    for (int r = 0; r < 8; ++r)
        out[(long)r * NS] = acc[r];
}

// ---------------------------------------------------------------------------
// K3a: chunk-local scan (h starts at 0), store end-of-chunk state.
// One lane owns one d: h[0..127], a[0..127] live in VGPRs.
// Bt rows are wave-uniform -> scalar-cache loads; x loads are coalesced b32.
// carry layout: [b, c, d, n]
// ---------------------------------------------------------------------------
__global__ void scan_chunk_local_kernel(const float* __restrict__ X,
                                        const float* __restrict__ Bt,
                                        const float* __restrict__ A_log,
                                        float* __restrict__ carry) {
    const int b = blockIdx.z;
    const int c = blockIdx.y;
    const int d = blockIdx.x * 256 + threadIdx.x;

    float a[NS], h[NS];
#pragma unroll
    for (int n = 0; n < NS; ++n) {
        a[n] = -expf(A_log[(long)d * NS + n]);
        h[n] = 0.0f;
    }

    const float* __restrict__ xp = X  + ((long)b * SEQ + (long)c * CHUNK) * DM + d;
    const float* __restrict__ bp = Bt + ((long)b * SEQ + (long)c * CHUNK) * NS;

    for (int t = 0; t < CHUNK; ++t) {
        __builtin_prefetch(xp + (long)(t + 8) * DM, 0, 1);   // global_prefetch_b8
        const float x = xp[(long)t * DM];
        const float* __restrict__ bt = bp + t * NS;          // wave-uniform row
#pragma unroll
        for (int n = 0; n < NS; ++n)
            h[n] = fmaf(a[n], h[n], bt[n] * x);
    }

    float* __restrict__ cp = carry + (((long)b * NCHUNK + c) * DM + d) * NS;
#pragma unroll
    for (int n = 0; n < NS; ++n) cp[n] = h[n];
}

// ---------------------------------------------------------------------------
// K3b: combine chunk carries into exclusive-prefix start states (in place).
// One thread per (b,d,n); a^CHUNK via 8 squarings (CHUNK = 2^8).
// ---------------------------------------------------------------------------
__global__ void scan_chunk_combine_kernel(const float* __restrict__ A_log,
                                          float* __restrict__ carry) {
    const long idx = (long)blockIdx.x * 256 + threadIdx.x;   // over B*DM*NS
    const int  n   = (int)(idx & (NS - 1));
    const long dn  = idx >> 7;
    const int  d   = (int)(dn & (DM - 1));
    const int  b   = (int)(dn >> 10);

    float a = -expf(A_log[(long)d * NS + n]);
    float ac = a;
#pragma unroll
    for (int i = 0; i < 8; ++i) ac *= ac;                    // a^256

    float* __restrict__ base = carry + ((long)b * NCHUNK) * DM * NS
                                     + (long)d * NS + n;
    float hpre = 0.0f;
    for (int c = 0; c < NCHUNK; ++c) {
        const float local = base[(long)c * DM * NS];
        base[(long)c * DM * NS] = hpre;                      // exclusive prefix
        hpre = fmaf(ac, hpre, local);
    }
}

// ---------------------------------------------------------------------------
// K3c: re-run each chunk from its corrected start state, emit y (pre-norm).
// y[b,t,d] = sum_n h[n] + D[d]*x
// ---------------------------------------------------------------------------
__global__ void scan_chunk_output_kernel(const float* __restrict__ X,
                                         const float* __restrict__ Bt,
                                         const float* __restrict__ A_log,
                                         const float* __restrict__ carry,
                                         const float* __restrict__ Dp,
                                         float* __restrict__ Y) {
    const int b = blockIdx.z;
    const int c = blockIdx.y;
    const int d = blockIdx.x * 256 + threadIdx.x;

    const float* __restrict__ cp = carry + (((long)b * NCHUNK + c) * DM + d) * NS;
    float a[NS], h[NS];
#pragma unroll
    for (int n = 0; n < NS; ++n) {
        a[n] = -expf(A_log[(long)d * NS + n]);
        h[n] = cp[n];
    }
    const float Dd = Dp[d];

    const float* __restrict__ xp = X  + ((long)b * SEQ + (long)c * CHUNK) * DM + d;
    const float* __restrict__ bp = Bt + ((long)b * SEQ + (long)c * CHUNK) * NS;
    float* __restrict__ yp = Y + ((long)b * SEQ + (long)c * CHUNK) * DM + d;

    for (int t = 0; t < CHUNK; ++t) {
        __builtin_prefetch(xp + (long)(t + 8) * DM, 0, 1);
        const float x = xp[(long)t * DM];
        const float* __restrict__ bt = bp + t * NS;          // wave-uniform row
        float s = 0.0f;
#pragma unroll
        for (int n = 0; n < NS; ++n) {
            h[n] = fmaf(a[n], h[n], bt[n] * x);
            s += h[n];
        }
        yp[(long)t * DM] = fmaf(Dd, x, s);
    }
}

// ---------------------------------------------------------------------------
// K4: in-place RMSNorm over D=1024 per (b,t) row. eps = FLT_EPSILON (torch/jnp
// float32 eps = 1.1920929e-7). Block 256 = 8 wave32; float4 per thread.
// ---------------------------------------------------------------------------
__global__ void rmsnorm_kernel(float* __restrict__ Y,
                               const float* __restrict__ w) {
    const long row = blockIdx.x;                             // b*SEQ + s
    float4* __restrict__ yr = (float4*)(Y + row * DM);

    float4 v = yr[threadIdx.x];
    float ss = v.x * v.x + v.y * v.y + v.z * v.z + v.w * v.w;

    // wave32 reduce
#pragma unroll
    for (int off = 16; off > 0; off >>= 1)
        ss += __shfl_down(ss, off);

    __shared__ float red[8];
    const int lane = threadIdx.x & 31;
    const int wv   = threadIdx.x >> 5;
    if (lane == 0) red[wv] = ss;
    __syncthreads();

    float total = 0.0f;
#pragma unroll
    for (int i = 0; i < 8; ++i) total += red[i];

    const float inv = rsqrtf(total * (1.0f / (float)DM) + 1.1920929e-07f);
    const float4 wvv = ((const float4*)w)[threadIdx.x];
    v.x *= inv * wvv.x;
    v.y *= inv * wvv.y;
    v.z *= inv * wvv.z;
    v.w *= inv * wvv.w;
    yr[threadIdx.x] = v;
}

// ---------------------------------------------------------------------------
extern "C" void kernel_launch(void* const* d_in, const int* in_sizes, int n_in,
                              void* d_out, int out_size, void* d_ws, size_t ws_size,
                              hipStream_t stream) {
    const int*   tokens = (const int*)  d_in[0];   // [B, S]
    const float* emb    = (const float*)d_in[1];   // [VOCAB, DM]
    const float* A_log  = (const float*)d_in[2];   // [DM, NS]
    const float* B_w    = (const float*)d_in[3];   // [NS, DM]
    const float* D_p    = (const float*)d_in[4];   // [DM]
    const float* norm_w = (const float*)d_in[5];   // [DM]
    float* Y = (float*)d_out;                      // [B, S, DM]

    // workspace layout
    char* ws = (char*)d_ws;
    float* X     = (float*)ws;                                        // 64 MB
    float* Bt    = (float*)(ws + (size_t)BATCH * SEQ * DM * 4);       //  8 MB
    float* carry = (float*)(ws + (size_t)BATCH * SEQ * DM * 4
                               + (size_t)BATCH * SEQ * NS * 4);       // 32 MB

    // K1: gather
    gather_embed_kernel<<<BATCH * SEQ, 256, 0, stream>>>(tokens, emb, X);

    // K2: Bt = X @ B_w^T via v_wmma_f32_16x16x4_f32
    bproj_wmma_kernel<<<(BATCH * SEQ) / 16, 256, 0, stream>>>(X, B_w, Bt);

    // K3: chunked diagonal scan
    dim3 scan_grid(DM / 256, NCHUNK, BATCH);
    scan_chunk_local_kernel<<<scan_grid, 256, 0, stream>>>(X, Bt, A_log, carry);
    scan_chunk_combine_kernel<<<(BATCH * DM * NS) / 256, 256, 0, stream>>>(A_log, carry);
    scan_chunk_output_kernel<<<scan_grid, 256, 0, stream>>>(X, Bt, A_log, carry, D_p, Y);

    // K4: RMSNorm in place on d_out
    rmsnorm_kernel<<<BATCH * SEQ, 256, 0, stream>>>(Y, norm_w);
}